// FractalConsciousnessProcessor_47699906789699
// MI455X (gfx1250) — compile-verified
//
#include <hip/hip_runtime.h>

// ---------------------------------------------------------------------------
// FractalConsciousnessProcessor for MI455X (gfx1250, wave32)
//   kernel 1: sum-over-S (1 GB stream) + softmax + logistic modulation -> psi0
//   kernel 2: one block per batch row; psi row resident in LDS for all steps;
//             fractal field via chained V_WMMA_F32_16X16X4_F32 (K=20 in 5x K=4)
// ---------------------------------------------------------------------------

#define EDIM   1024
#define TPB    256
#define WAVES  (TPB / 32)
#define NLAM   20

#define PHASEC   0.7854f
#define CHAOTIC  3.9f
#define DTC      0.01f
#define OMEGA    2.75f
#define DMIN     0.01f
#define DMAX     10.0f
#define MAXF     10.0f
#define EPSV     1e-10f
#define INV2PI   0.15915494309189535f
#define LOG2E    1.4426950408889634f
#define LN2      0.6931471805599453f
#define LN_EDIM  6.931471805599453f   // ln(1024)

typedef float v2f __attribute__((ext_vector_type(2)));
typedef float v4f __attribute__((ext_vector_type(4)));
typedef float v8f __attribute__((ext_vector_type(8)));

// v_sin_f32 takes revolutions: computes sin(2*pi*x). Valid |x| <= 256.
__device__ __forceinline__ float sin2pi(float x) { return __builtin_amdgcn_sinf(x); }
__device__ __forceinline__ float exp2h(float x)  { return __builtin_amdgcn_exp2f(x); }
__device__ __forceinline__ float log2h(float x)  { return __builtin_amdgcn_logf(x); }

// ---------------------------------------------------------------------------
// Kernel 0: zero the fci accumulation slots (atomically accumulated later)
// ---------------------------------------------------------------------------
__global__ void fcp_zero_fci(float* __restrict__ out, const int* __restrict__ nsp, int B) {
    int steps = *nsp;
    size_t off = (size_t)(2 + steps) * (size_t)B * EDIM;
    int t = (int)threadIdx.x;
    if (t < steps) out[off + t] = 0.0f;
}

// ---------------------------------------------------------------------------
// Kernel 1: psi0 = normalize(softmax(sum_s input) * logistic_mod(x0))
//           (softmax denominator cancels in the final normalization)
// Writes psi0 into out[0 .. B*E) (the psi_final slot, overwritten at the end).
// ---------------------------------------------------------------------------
__global__ __launch_bounds__(TPB)
void fcp_init(const float* __restrict__ in, const float* __restrict__ x0,
              float* __restrict__ out, int S) {
    __shared__ float red[TPB];
    const int b = (int)blockIdx.x, t = (int)threadIdx.x;

    const v4f* row = (const v4f*)(in + (size_t)b * (size_t)S * (size_t)EDIM);
    v4f acc = {0.f, 0.f, 0.f, 0.f};
    for (int s = 0; s < S; ++s) {
        if (s + 8 < S) __builtin_prefetch(&row[(size_t)(s + 8) * (EDIM / 4) + t], 0, 0);
        v4f v = row[(size_t)s * (EDIM / 4) + t];
        acc += v;
    }

    // block max over E (softmax stabilization)
    float lm = fmaxf(fmaxf(acc.x, acc.y), fmaxf(acc.z, acc.w));
    red[t] = lm; __syncthreads();
    for (int s = TPB / 2; s > 0; s >>= 1) {
        if (t < s) red[t] = fmaxf(red[t], red[t + s]);
        __syncthreads();
    }
    const float M = red[0]; __syncthreads();

    float a4[4] = {acc.x, acc.y, acc.z, acc.w};
    float p[4]; float psum = 0.f;
#pragma unroll
    for (int j = 0; j < 4; ++j) {
        int e = 4 * t + j;
        float ex = exp2h((a4[j] - M) * LOG2E);
        float x = x0[(size_t)b * EDIM + e] * 0.5f + 0.25f;
#pragma unroll
        for (int it = 0; it < 10; ++it) x = CHAOTIC * x * (1.0f - x);
        float mod = 0.5f + 0.5f * sin2pi(x);
        p[j] = ex * mod; psum += p[j];
    }
    red[t] = psum; __syncthreads();
    for (int s = TPB / 2; s > 0; s >>= 1) {
        if (t < s) red[t] += red[t + s];
        __syncthreads();
    }
    const float inv = 1.0f / red[0];
#pragma unroll
    for (int j = 0; j < 4; ++j) out[(size_t)b * EDIM + 4 * t + j] = p[j] * inv;
}

// ---------------------------------------------------------------------------
// Kernel 2: 50-step evolution; one block per batch row, psi row in LDS.
// Field: D = A x B + C with A[m,k] = sin(2pi*psi[e_m]*(k+1) + phase),
//        B[k,n] = lambda[k] for all n  => all D columns identical = field.
// A 16x4 f32 layout: lanes 0-15 K={0,1}, lanes 16-31 K={2,3} (VGPR0/1).
// C/D 16x16 f32 layout: VGPR r: lanes 0-15 M=r, lanes 16-31 M=r+8.
// ---------------------------------------------------------------------------
__global__ __launch_bounds__(TPB)
void fcp_evolve(const float* __restrict__ lam, const int* __restrict__ nsp,
                float* __restrict__ out, int B) {
    __shared__ float psi[EDIM];
    __shared__ float fld[EDIM];
    __shared__ float red[TPB];
    __shared__ float red2[TPB];

    const int b = (int)blockIdx.x, t = (int)threadIdx.x;
    const int lane = t & 31, wave = t >> 5;
    const int m = lane & 15;            // matrix row this lane owns
    const int kh = (lane >> 4) * 2;     // K offset of this lane-half (0 or 2)
    const int steps = *nsp;

    const size_t BE = (size_t)B * EDIM;
    float* traj  = out + 2 * BE;                      // (steps, B, E)
    float* fci   = out + (2 + (size_t)steps) * BE;    // (steps,)
    float* dlast = fci + steps;                       // (B,)

    // load psi0 (staged in out[0..BE) by fcp_init)
#pragma unroll
    for (int j = 0; j < 4; ++j) psi[4 * t + j] = out[(size_t)b * EDIM + 4 * t + j];

    // preload this lane's lambda pairs for each K=4 chunk
    v2f blam[5];
#pragma unroll
    for (int c = 0; c < 5; ++c) { blam[c].x = lam[4 * c + kh]; blam[c].y = lam[4 * c + kh + 1]; }
    __syncthreads();

    float diff = 0.0f;
    for (int step = 0; step < steps; ++step) {
        const float cphase = (PHASEC + OMEGA * (step * DTC)) * INV2PI;

        // ---- fractal field via WMMA (uniform control flow, EXEC all ones) ----
        for (int tile = wave; tile < EDIM / 16; tile += WAVES) {
            const int base = tile * 16;
            const float p = psi[base + m];
            v8f acc = {0.f, 0.f, 0.f, 0.f, 0.f, 0.f, 0.f, 0.f};
#pragma unroll
            for (int c = 0; c < 5; ++c) {
                const float k1 = (float)(4 * c + kh + 1);
                v2f a;
                a.x = sin2pi(p * k1 + cphase);
                a.y = sin2pi(p * (k1 + 1.0f) + cphase);
                acc = __builtin_amdgcn_wmma_f32_16x16x4_f32(
                          false, a, false, blam[c], (short)0, acc, false, false);
            }
            // all D columns identical: lane 0 holds M=0..7, lane 16 holds M=8..15
            if (m == 0) {
                const int off = base + (lane >> 4) * 8;
#pragma unroll
                for (int r = 0; r < 8; ++r) {
                    float f = acc[r];
                    fld[off + r] = fminf(fmaxf(f, -MAXF), MAXF);
                }
            }
        }
        __syncthreads();

        // ---- diffusion coefficient from field variance over E ----
        float s1 = 0.f, s2 = 0.f;
#pragma unroll
        for (int j = 0; j < 4; ++j) { float f = fld[4 * t + j]; s1 += f; s2 += f * f; }
        red[t] = s1; red2[t] = s2; __syncthreads();
        for (int s = TPB / 2; s > 0; s >>= 1) {
            if (t < s) { red[t] += red[t + s]; red2[t] += red2[t + s]; }
            __syncthreads();
        }
        const float mean = red[0] * (1.0f / EDIM);
        const float var  = red2[0] * (1.0f / EDIM) - mean * mean;
        __syncthreads();
        const float sig = 1.0f / (1.0f + exp2h(-var * LOG2E));
        diff = DMIN + (DMAX - DMIN) * sig;

        // ---- circular divergence + laplacian update ----
        float nv[4]; float psum = 0.f;
#pragma unroll
        for (int j = 0; j < 4; ++j) {
            const int e  = 4 * t + j;
            const int ep = (e + 1) & (EDIM - 1);
            const int em = (e - 1) & (EDIM - 1);
            const float dv = 0.5f * (fld[ep] * psi[ep] - fld[em] * psi[em]);
            const float lp = psi[ep] - 2.0f * psi[e] + psi[em];
            float x = psi[e] + DTC * (-dv + diff * lp);
            nv[j] = fmaxf(x, EPSV); psum += nv[j];
        }
        red[t] = psum; __syncthreads();
        for (int s = TPB / 2; s > 0; s >>= 1) {
            if (t < s) red[t] += red[t + s];
            __syncthreads();
        }
        const float inv = 1.0f / red[0];
        __syncthreads();

        // ---- commit psi, write trajectory (NT), entropy partials ----
        float hp = 0.f;
        v4f tv;
        {
            float v0 = nv[0] * inv, v1 = nv[1] * inv, v2 = nv[2] * inv, v3 = nv[3] * inv;
            psi[4 * t + 0] = v0; psi[4 * t + 1] = v1;
            psi[4 * t + 2] = v2; psi[4 * t + 3] = v3;
            tv.x = v0; tv.y = v1; tv.z = v2; tv.w = v3;
            hp += v0 * log2h(v0 + EPSV); hp += v1 * log2h(v1 + EPSV);
            hp += v2 * log2h(v2 + EPSV); hp += v3 * log2h(v3 + EPSV);
        }
        v4f* trow = (v4f*)(traj + (size_t)step * BE + (size_t)b * EDIM);
        __builtin_nontemporal_store(tv, &trow[t]);

        red[t] = hp; __syncthreads();
        for (int s = TPB / 2; s > 0; s >>= 1) {
            if (t < s) red[t] += red[t + s];
            __syncthreads();
        }
        if (t == 0) {
            const float H = -red[0] * LN2;               // log2 -> ln
            atomicAdd(&fci[step], (1.0f - H * (1.0f / LN_EDIM)) * (1.0f / (float)B));
        }
        __syncthreads();
    }

    // ---- finals: psi_final, field_last, diff_last ----
#pragma unroll
    for (int j = 0; j < 4; ++j) {
        const int e = 4 * t + j;
        out[(size_t)b * EDIM + e]      = psi[e];
        out[BE + (size_t)b * EDIM + e] = fld[e];
    }
    if (t == 0) dlast[b] = diff;
}

// ---------------------------------------------------------------------------
extern "C" void kernel_launch(void* const* d_in, const int* in_sizes, int n_in,
                              void* d_out, int out_size, void* d_ws, size_t ws_size,
                              hipStream_t stream) {
    (void)n_in; (void)out_size; (void)d_ws; (void)ws_size;
    const float* input = (const float*)d_in[0];   // (B, S, E) f32
    const float* x0    = (const float*)d_in[1];   // (B, E)    f32
    const float* lam   = (const float*)d_in[2];   // (20,)     f32
    const int*   nstep = (const int*)d_in[3];     // scalar    i32 (read on device)
    float* out = (float*)d_out;

    const int BE = in_sizes[1];
    const int B  = BE / EDIM;
    const int S  = in_sizes[0] / BE;

    fcp_zero_fci<<<1, TPB, 0, stream>>>(out, nstep, B);
    fcp_init<<<B, TPB, 0, stream>>>(input, x0, out, S);
    fcp_evolve<<<B, TPB, 0, stream>>>(lam, nstep, out, B);
}